// GNN_12000138625510
// MI455X (gfx1250) — compile-verified
//
#include <hip/hip_runtime.h>

// GIN forward, 3 layers: agg[i] = sum_{(s->i)} h[s];  h' = ((1+eps)*h + agg) @ W + b
// N_NODES = 100000 (= 6250 * 16), N_EDGES = 3.2M, DIM = 128.

typedef __attribute__((ext_vector_type(2))) float v2f;
typedef __attribute__((ext_vector_type(8))) float v8f;

#define HID 128
#define LDA 132   // padded LDS row stride for the A tile (bank-conflict avoidance)

// ---------------------------------------------------------------- zero agg
__global__ __launch_bounds__(256) void gin_zero_kernel(float4* __restrict__ p, int n4) {
  int i = blockIdx.x * blockDim.x + threadIdx.x;
  if (i < n4) p[i] = make_float4(0.f, 0.f, 0.f, 0.f);
}

// ---------------------------------------------------------------- edge scatter-add
// one wave (32 lanes) per edge; each lane moves 4 channels (float4)
__global__ __launch_bounds__(256) void gin_scatter_kernel(
    const float* __restrict__ h, const int* __restrict__ esrc,
    const int* __restrict__ edst, float* __restrict__ agg, int n_edges) {
  int gid  = blockIdx.x * blockDim.x + threadIdx.x;
  int e    = gid >> 5;                 // wave-uniform edge id
  if (e >= n_edges) return;
  int lane = threadIdx.x & 31;
  int s = __builtin_amdgcn_readfirstlane(esrc[e]);
  int d = __builtin_amdgcn_readfirstlane(edst[e]);
  const float4 v = ((const float4*)(h + (size_t)s * HID))[lane];
  float* a = agg + (size_t)d * HID + lane * 4;
  atomicAdd(a + 0, v.x);
  atomicAdd(a + 1, v.y);
  atomicAdd(a + 2, v.z);
  atomicAdd(a + 3, v.w);
}

// ---------------------------------------------------------------- fused update GEMM
// block = 256 threads = 8 waves; block handles rows [16*tile_m, 16*tile_m+16),
// wave w handles output columns [16*w, 16*w+16).  In-place safe (reads its own
// 16 rows into LDS before writing them back).
__global__ __launch_bounds__(256) void gin_update_kernel(
    const float* __restrict__ h, const float* __restrict__ agg,
    const float* __restrict__ W, const float* __restrict__ bias,
    const float* __restrict__ epsp, float* __restrict__ out) {
  extern __shared__ float sA[];        // [16][LDA] fused A tile

  const int tid    = threadIdx.x;
  const int tile_m = blockIdx.x;
  const float scale = 1.0f + epsp[0];

  // stage fused A tile: sA[r][c] = (1+eps)*h[row,c] + agg[row,c]
  for (int i = tid; i < 16 * HID; i += 256) {
    int r = i >> 7;          // 0..15
    int c = i & (HID - 1);   // 0..127
    size_t g = (size_t)(tile_m * 16 + r) * HID + c;
    sA[r * LDA + c] = scale * h[g] + agg[g];
  }
  __syncthreads();

  const int wave     = tid >> 5;       // 0..7  -> N tile
  const int lane     = tid & 31;
  const int lane_lo  = lane & 15;
  const int hi       = lane >> 4;      // 0/1 selects K pair {0,1} vs {2,3}
  const int n_global = wave * 16 + lane_lo;

  const float* sArow = sA + lane_lo * LDA;     // A frag: row = lane&15

  v8f acc = {};
  #pragma unroll
  for (int k0 = 0; k0 < HID; k0 += 4) {
    const int ka = k0 + hi * 2;
    v2f a, b;
    a.x = sArow[ka];
    a.y = sArow[ka + 1];
    b.x = W[(size_t)ka       * HID + n_global];
    b.y = W[(size_t)(ka + 1) * HID + n_global];
    // D = A(16x4,f32) x B(4x16,f32) + C  -> v_wmma_f32_16x16x4_f32
    acc = __builtin_amdgcn_wmma_f32_16x16x4_f32(
        /*neg_a=*/false, a, /*neg_b=*/false, b,
        /*c_mod=*/(short)0, acc, /*reuse_a=*/false, /*reuse_b=*/false);
  }

  const float bv = bias[n_global];
  #pragma unroll
  for (int v = 0; v < 8; ++v) {
    // C/D layout: VGPR v -> row v + 8*hi, col = lane&15
    size_t g = (size_t)(tile_m * 16 + v + 8 * hi) * HID + n_global;
    out[g] = acc[v] + bv;
  }
}

// ---------------------------------------------------------------- launcher
extern "C" void kernel_launch(void* const* d_in, const int* in_sizes, int n_in,
                              void* d_out, int out_size, void* d_ws, size_t ws_size,
                              hipStream_t stream) {
  const float* feats = (const float*)d_in[0];
  const int*   esrc  = (const int*)d_in[1];
  const int*   edst  = (const int*)d_in[2];
  const float* Wl[3]  = {(const float*)d_in[3], (const float*)d_in[6], (const float*)d_in[9]};
  const float* bl[3]  = {(const float*)d_in[4], (const float*)d_in[7], (const float*)d_in[10]};
  const float* epl[3] = {(const float*)d_in[5], (const float*)d_in[8], (const float*)d_in[11]};

  float* out = (float*)d_out;
  float* agg = (float*)d_ws;           // 100000*128*4 = 51.2 MB scratch

  const int n_edges = in_sizes[1];
  const int n_nodes = out_size / HID;  // 100000
  const int n4      = n_nodes * (HID / 4);
  const int zero_blocks    = (n4 + 255) / 256;
  const int scatter_blocks = (n_edges + 7) / 8;      // 8 waves (edges) per block
  const int gemm_blocks    = n_nodes / 16;           // 6250, exact
  const size_t smem        = (size_t)16 * LDA * sizeof(float);

  const float* hin = feats;
  for (int l = 0; l < 3; ++l) {
    gin_zero_kernel<<<zero_blocks, 256, 0, stream>>>((float4*)agg, n4);
    gin_scatter_kernel<<<scatter_blocks, 256, 0, stream>>>(hin, esrc, edst, agg, n_edges);
    gin_update_kernel<<<gemm_blocks, 256, smem, stream>>>(hin, agg, Wl[l], bl[l], epl[l], out);
    hin = out;                          // layers 1,2 run in-place on d_out
  }
}